// MSSSIM_62818191671494
// MI455X (gfx1250) — compile-verified
//
#include <hip/hip_runtime.h>
#include <cmath>

typedef __attribute__((ext_vector_type(2))) float v2f;
typedef __attribute__((ext_vector_type(8))) float v8f;

#define WIN 11
#define PLANES 96          // B*C = 32*3

// Gaussian window, sigma=1.5, normalized (matches reference _make_window)
__constant__ float c_g[WIN] = {
    0.00102838f, 0.00759874f, 0.03600077f, 0.10936069f, 0.21300553f,
    0.26601172f,
    0.21300553f, 0.10936069f, 0.03600077f, 0.00759874f, 0.00102838f
};

__device__ __forceinline__ v8f wmma4(v2f a, v2f b, v8f c) {
    return __builtin_amdgcn_wmma_f32_16x16x4_f32(false, a, false, b, (short)0, c,
                                                 false, false);
}

__device__ __forceinline__ void atomicMaxFloat(float* addr, float v) {
    if (v >= 0.f) atomicMax((int*)addr, __float_as_int(v));
    else          atomicMin((unsigned int*)addr, __float_as_uint(v));
}
__device__ __forceinline__ void atomicMinFloat(float* addr, float v) {
    if (v <= 0.f) atomicMax((unsigned int*)addr, __float_as_uint(v));
    else          atomicMin((int*)addr, __float_as_int(v));
}

// ---------------------------------------------------------------------------
// init: zero the 10 sum slots, set max=-inf / min=+inf for 5 levels
// ---------------------------------------------------------------------------
__global__ void msssim_init_kernel(float* s) {
    int i = threadIdx.x;
    if (i < 10)      s[i] = 0.f;
    else if (i < 15) s[i] = -INFINITY;   // vmax[l] at s[10+l]
    else if (i < 20) s[i] =  INFINITY;   // vmin[l] at s[15+l]
}

// ---------------------------------------------------------------------------
// global max/min of level-0 img1 (dynamic-range L computation)
// ---------------------------------------------------------------------------
__global__ __launch_bounds__(256)
void msssim_range_kernel(const float* __restrict__ in, int n,
                         float* vmax, float* vmin) {
    __shared__ float sMax[256], sMin[256];
    float lmax = -INFINITY, lmin = INFINITY;
    for (int i = blockIdx.x * blockDim.x + threadIdx.x; i < n;
         i += gridDim.x * blockDim.x) {
        float v = in[i];
        lmax = fmaxf(lmax, v);
        lmin = fminf(lmin, v);
    }
    sMax[threadIdx.x] = lmax; sMin[threadIdx.x] = lmin;
    __syncthreads();
    for (int s = 128; s > 0; s >>= 1) {
        if (threadIdx.x < s) {
            sMax[threadIdx.x] = fmaxf(sMax[threadIdx.x], sMax[threadIdx.x + s]);
            sMin[threadIdx.x] = fminf(sMin[threadIdx.x], sMin[threadIdx.x + s]);
        }
        __syncthreads();
    }
    if (threadIdx.x == 0) {
        atomicMaxFloat(vmax, sMax[0]);
        atomicMinFloat(vmin, sMin[0]);
    }
}

// ---------------------------------------------------------------------------
// 2x2 avg pool; optionally tracks max/min of the *pooled* output, which is
// exactly the i1 tensor the next SSIM level sees.
// ---------------------------------------------------------------------------
__global__ __launch_bounds__(256)
void msssim_pool_kernel(const float* __restrict__ in, float* __restrict__ out,
                        int Hout, int Wout, int doRange,
                        float* vmax, float* vmin) {
    __shared__ float sMax[256], sMin[256];
    const int Hin = Hout * 2, Win_ = Wout * 2;
    const int nOut = PLANES * Hout * Wout;
    float lmax = -INFINITY, lmin = INFINITY;
    for (int i = blockIdx.x * blockDim.x + threadIdx.x; i < nOut;
         i += gridDim.x * blockDim.x) {
        int p = i / (Hout * Wout);
        int r = (i / Wout) % Hout;
        int c = i % Wout;
        const float* b = in + ((size_t)p * Hin + 2 * r) * Win_ + 2 * c;
        float v = 0.25f * (b[0] + b[1] + b[Win_] + b[Win_ + 1]);
        out[i] = v;
        lmax = fmaxf(lmax, v);
        lmin = fminf(lmin, v);
    }
    if (doRange) {
        sMax[threadIdx.x] = lmax; sMin[threadIdx.x] = lmin;
        __syncthreads();
        for (int s = 128; s > 0; s >>= 1) {
            if (threadIdx.x < s) {
                sMax[threadIdx.x] = fmaxf(sMax[threadIdx.x], sMax[threadIdx.x + s]);
                sMin[threadIdx.x] = fminf(sMin[threadIdx.x], sMin[threadIdx.x + s]);
            }
            __syncthreads();
        }
        if (threadIdx.x == 0) {
            atomicMaxFloat(vmax, sMax[0]);
            atomicMinFloat(vmin, sMin[0]);
        }
    }
}

// ---------------------------------------------------------------------------
// Fused SSIM level kernel.
// One wave computes one 16x16 tile of the (H-10)x(W-10) conv-output plane.
// Both separable conv passes run on the fp32 matrix pipe:
//   horizontal: D = Q(16x28) * G_band(28x16)  -> 7 x V_WMMA_F32_16X16X4_F32
//   vertical:   D = G_band(16x28) * H(28x16)  -> 7 x V_WMMA_F32_16X16X4_F32
// applied to the 5 quantities {i1, i2, i1^2, i2^2, i1*i2}.
// The band fragments g[k-n] are identical for B(horizontal) and A(vertical),
// so they are loaded once per chunk from a zero-extended LDS table, with no
// EXEC-mask predication anywhere in the hot path (WMMA needs EXEC all ones).
// f32 16x16x4 lane layout (ISA 7.12.2):
//   A[m][k]: lane<16 -> m=lane, regs {k=0,1}; lane>=16 -> m=lane-16, {k=2,3}
//   B[k][n]: lane<16 -> n=lane, regs {k=0,1}; lane>=16 -> n=lane-16, {k=2,3}
//   D[m][n]: reg v, lane<16 -> m=v; lane>=16 -> m=v+8
// ---------------------------------------------------------------------------
__global__ __launch_bounds__(128)
void msssim_ssim_kernel(const float* __restrict__ i1,
                        const float* __restrict__ i2,
                        int H, int W,
                        const float* __restrict__ vmax,
                        const float* __restrict__ vmin,
                        float* __restrict__ ssim_sum,
                        float* __restrict__ cs_sum,
                        int totalTiles, int tilesX, int tilesY) {
    __shared__ float sGx[64];                // taps at [16..26], zeros elsewhere
    __shared__ float sIn1[4][26][28];        // per-wave input tiles (padded)
    __shared__ float sIn2[4][26][28];
    __shared__ float sH[4][5][16][28];       // horizontal result, transposed [n][k]
    __shared__ float sRed[128][2];

    const int tid   = threadIdx.x;
    const int wave  = tid >> 5;
    const int lane  = tid & 31;
    const int n     = lane & 15;
    const int khalf = lane >> 4;
    const int OH = H - 10, OW = W - 10;

    if (tid < 64) sGx[tid] = (tid >= 16 && tid < 16 + WIN) ? c_g[tid - 16] : 0.f;
    __syncthreads();

    int tIdx = blockIdx.x * 4 + wave;
    const float tileValid = (tIdx < totalTiles) ? 1.f : 0.f;
    if (tIdx >= totalTiles) tIdx = totalTiles - 1;   // clamp, keep wave working

    const int tilesPerPlane = tilesX * tilesY;
    const int plane = tIdx / tilesPerPlane;
    const int rem   = tIdx % tilesPerPlane;
    const int row0  = (rem / tilesX) * 16;
    const int col0  = (rem % tilesX) * 16;
    const float* p1 = i1 + (size_t)plane * H * W;
    const float* p2 = i2 + (size_t)plane * H * W;

    // Stage input tiles (26x28, edge-clamped -> finite everywhere).
    // Fixed trip count + clamped index keeps EXEC fully populated.
    #pragma unroll
    for (int i = 0; i < 23; ++i) {
        int idx = lane + i * 32;
        idx = min(idx, 26 * 28 - 1);
        int r  = idx / 28, cc = idx % 28;
        int gr = min(row0 + r, H - 1);
        int gc = min(col0 + cc, W - 1);
        size_t o = (size_t)gr * W + gc;
        sIn1[wave][r][cc] = p1[o];
        sIn2[wave][r][cc] = p2[o];
    }

    // Band fragments per K-chunk: bg[c] = { g[kk-n], g[kk+1-n] } (zero-padded
    // table => single unpredicated ds_load per element, index in [1,44]).
    v2f bg[7];
    #pragma unroll
    for (int c = 0; c < 7; ++c) {
        const int kk = 4 * c + 2 * khalf;
        bg[c].x = sGx[16 + kk - n];
        bg[c].y = sGx[16 + kk + 1 - n];
    }

    // dynamic range -> C1/C2 (reference val_range=None path)
    const float mxv = vmax[0], mnv = vmin[0];
    const float L  = ((mxv > 128.f) ? 255.f : 1.f) - ((mnv < -0.5f) ? -1.f : 0.f);
    const float C1 = (0.01f * L) * (0.01f * L);
    const float C2 = (0.03f * L) * (0.03f * L);

    // -------- horizontal pass: rows [0..15] and [10..25] --------
    #pragma unroll
    for (int half = 0; half < 2; ++half) {
        const int r0 = half * 10;
        v8f acc[5];
        #pragma unroll
        for (int q = 0; q < 5; ++q)
            #pragma unroll
            for (int e = 0; e < 8; ++e) acc[q][e] = 0.f;

        #pragma unroll
        for (int c = 0; c < 7; ++c) {
            const int kk = 4 * c + 2 * khalf;
            float x0 = sIn1[wave][r0 + n][kk], x1 = sIn1[wave][r0 + n][kk + 1];
            float y0 = sIn2[wave][r0 + n][kk], y1 = sIn2[wave][r0 + n][kk + 1];
            v2f a;
            a.x = x0;      a.y = x1;      acc[0] = wmma4(a, bg[c], acc[0]);
            a.x = y0;      a.y = y1;      acc[1] = wmma4(a, bg[c], acc[1]);
            a.x = x0 * x0; a.y = x1 * x1; acc[2] = wmma4(a, bg[c], acc[2]);
            a.x = y0 * y0; a.y = y1 * y1; acc[3] = wmma4(a, bg[c], acc[3]);
            a.x = x0 * y0; a.y = x1 * y1; acc[4] = wmma4(a, bg[c], acc[4]);
        }
        // store transposed: sH[q][n][row]
        #pragma unroll
        for (int v = 0; v < 8; ++v) {
            const int hr = r0 + v + khalf * 8;
            #pragma unroll
            for (int q = 0; q < 5; ++q) sH[wave][q][n][hr] = acc[q][v];
        }
    }
    // zero K-padding columns 26,27 of sH
    #pragma unroll
    for (int q = 0; q < 5; ++q) sH[wave][q][n][26 + khalf] = 0.f;

    // -------- vertical pass: A = gaussian band, B = horizontal result --------
    v8f vac[5];
    #pragma unroll
    for (int q = 0; q < 5; ++q)
        #pragma unroll
        for (int e = 0; e < 8; ++e) vac[q][e] = 0.f;

    #pragma unroll
    for (int c = 0; c < 7; ++c) {
        const int kk = 4 * c + 2 * khalf;
        #pragma unroll
        for (int q = 0; q < 5; ++q) {
            v2f b;
            b.x = sH[wave][q][n][kk];
            b.y = sH[wave][q][n][kk + 1];
            vac[q] = wmma4(bg[c], b, vac[q]);
        }
    }

    // -------- per-pixel SSIM / CS + masked accumulation --------
    float ssA = 0.f, csA = 0.f;
    #pragma unroll
    for (int v = 0; v < 8; ++v) {
        const int orow = row0 + v + khalf * 8;
        const int ocol = col0 + n;
        float mu1 = vac[0][v], mu2 = vac[1][v];
        float x11 = vac[2][v], x22 = vac[3][v], x12 = vac[4][v];
        float mu1s = mu1 * mu1, mu2s = mu2 * mu2, m12 = mu1 * mu2;
        float s1 = x11 - mu1s, s2 = x22 - mu2s, s12 = x12 - m12;
        float cs = (2.f * s12 + C2) / (s1 + s2 + C2);
        float l  = (2.f * m12 + C1) / (mu1s + mu2s + C1);
        float ok = (orow < OH && ocol < OW) ? tileValid : 0.f;
        ssA += ok * (l * cs);
        csA += ok * cs;
    }

    // block reduction + one atomic per block
    sRed[tid][0] = ssA; sRed[tid][1] = csA;
    __syncthreads();
    for (int s = 64; s > 0; s >>= 1) {
        if (tid < s) {
            sRed[tid][0] += sRed[tid + s][0];
            sRed[tid][1] += sRed[tid + s][1];
        }
        __syncthreads();
    }
    if (tid == 0) {
        atomicAdd(ssim_sum, sRed[0][0]);
        atomicAdd(cs_sum,  sRed[0][1]);
    }
}

// ---------------------------------------------------------------------------
// finalize: means, pow-weighted product (reference combine)
// ---------------------------------------------------------------------------
__global__ void msssim_finalize_kernel(const float* s, float* out, int H0) {
    if (threadIdx.x == 0 && blockIdx.x == 0) {
        const float w[5] = {0.0448f, 0.2856f, 0.3001f, 0.2363f, 0.1333f};
        float prod = 1.f;
        for (int l = 0; l < 5; ++l) {
            int   Hl  = H0 >> l;
            float cnt = (float)PLANES * (float)(Hl - 10) * (float)(Hl - 10);
            float ms  = s[l]     / cnt;   // ssim mean
            float mc  = s[5 + l] / cnt;   // cs mean
            prod *= powf((l < 4) ? mc : ms, w[l]);
        }
        out[0] = prod;
    }
}

// ---------------------------------------------------------------------------
// host launcher
// ---------------------------------------------------------------------------
extern "C" void kernel_launch(void* const* d_in, const int* in_sizes, int n_in,
                              void* d_out, int out_size, void* d_ws, size_t ws_size,
                              hipStream_t stream) {
    const float* img1 = (const float*)d_in[0];
    const float* img2 = (const float*)d_in[1];
    float* out = (float*)d_out;
    float* ws  = (float*)d_ws;

    // derive H0 (square planes): PLANES * H0 * H0 == in_sizes[0]
    long long hw = (long long)in_sizes[0] / PLANES;
    int H0 = 1;
    while ((long long)H0 * H0 < hw) H0 <<= 1;

    // ws layout: [0..63] scalars, then pyramid buffers
    float* scal     = ws;
    float* ssim_sum = scal;          // [5]
    float* cs_sum   = scal + 5;      // [5]
    float* vmax     = scal + 10;     // [5]
    float* vmin     = scal + 15;     // [5]
    float* buf      = ws + 64;

    const float* i1p[5]; const float* i2p[5];
    float* i1b[5]; float* i2b[5];
    i1p[0] = img1; i2p[0] = img2;
    for (int l = 1; l < 5; ++l) {
        int Hl = H0 >> l;
        size_t nelem = (size_t)PLANES * Hl * Hl;
        i1b[l] = buf; buf += nelem;
        i2b[l] = buf; buf += nelem;
        i1p[l] = i1b[l]; i2p[l] = i2b[l];
    }

    msssim_init_kernel<<<1, 32, 0, stream>>>(scal);

    {   // level-0 dynamic range from img1
        int n = PLANES * H0 * H0;
        int blocks = (n + 255) / 256; if (blocks > 2048) blocks = 2048;
        msssim_range_kernel<<<blocks, 256, 0, stream>>>(img1, n, vmax, vmin);
    }

    for (int l = 0; l < 5; ++l) {
        int Hl = H0 >> l, Wl = Hl;
        int OH = Hl - 10, OW = Wl - 10;
        int tilesX = (OW + 15) / 16, tilesY = (OH + 15) / 16;
        int totalTiles = PLANES * tilesX * tilesY;
        int blocks = (totalTiles + 3) / 4;
        msssim_ssim_kernel<<<blocks, 128, 0, stream>>>(
            i1p[l], i2p[l], Hl, Wl, vmax + l, vmin + l,
            ssim_sum + l, cs_sum + l, totalTiles, tilesX, tilesY);

        if (l < 4) {
            int Ho = Hl / 2;
            int nOut = PLANES * Ho * Ho;
            int b2 = (nOut + 255) / 256; if (b2 > 2048) b2 = 2048;
            msssim_pool_kernel<<<b2, 256, 0, stream>>>(
                i1p[l], i1b[l + 1], Ho, Ho, 1, vmax + l + 1, vmin + l + 1);
            msssim_pool_kernel<<<b2, 256, 0, stream>>>(
                i2p[l], i2b[l + 1], Ho, Ho, 0, nullptr, nullptr);
        }
    }

    msssim_finalize_kernel<<<1, 32, 0, stream>>>(scal, out, H0);
}